// TextGraphSAGE_29583734734918
// MI455X (gfx1250) — compile-verified
//
#include <hip/hip_runtime.h>
#include <hip/hip_bf16.h>

typedef __attribute__((ext_vector_type(2))) float v2f;
typedef __attribute__((ext_vector_type(8))) float v8f;

#define N_NODES 50000
#define D_IN    768
#define H_DIM   128
#define C_DIM   4

// ---------------------------------------------------------------------------
// Degree: deg[dst] += 1 per edge
// ---------------------------------------------------------------------------
__global__ void sage_deg_kernel(const long long* __restrict__ eidx, long long E,
                                float* __restrict__ deg) {
    long long e = (long long)blockIdx.x * blockDim.x + threadIdx.x;
    if (e >= E) return;
    long long d = eidx[E + e];
    unsafeAtomicAdd(&deg[d], 1.0f);
}

// ---------------------------------------------------------------------------
// Fused GEMM1: p1 = x @ W1_l^T, r1 = x @ W1_r^T    (M=50000, K=768, N=128)
// One block = 4 waves. Wave w computes a 16x64 f32 output tile of either
// p1 (w=0,1) or r1 (w=2,3) using V_WMMA_F32_16X16X4_F32.
// For f32 16x16x4, A-from-rowmajor-X and B-from-rowmajor-W(=W^T) use the
// identical per-lane float2 addressing: base + (lane%16)*K + 2*(lane/16).
// ---------------------------------------------------------------------------
__global__ __launch_bounds__(128)
void sage_gemm1_wmma(const float* __restrict__ x,
                     const float* __restrict__ Wl,
                     const float* __restrict__ Wr,
                     float* __restrict__ p1,
                     float* __restrict__ r1) {
    const int wave = threadIdx.x >> 5;           // 0..3 (uniform per wave)
    const int lane = threadIdx.x & 31;
    const int m0   = blockIdx.x * 16;            // 3125 blocks, exact
    const float* W   = (wave < 2) ? Wl : Wr;
    float*       out = (wave < 2) ? p1 : r1;
    const int n0 = (wave & 1) * 64;

    const int mrow = lane & 15;                  // M (for A) / N (for B) index
    const int koff = (lane >> 4) << 1;           // 0 or 2

    const float* ap = x + (size_t)(m0 + mrow) * D_IN + koff;
    const float* bp = W + (size_t)(n0 + mrow) * D_IN + koff;

    v8f c0 = {}, c1 = {}, c2 = {}, c3 = {};

    #pragma unroll 4
    for (int k = 0; k < D_IN; k += 4) {
        v2f a  = *(const v2f*)(ap + k);
        v2f b0 = *(const v2f*)(bp + k);
        v2f b1 = *(const v2f*)(bp + 16 * D_IN + k);
        v2f b2 = *(const v2f*)(bp + 32 * D_IN + k);
        v2f b3 = *(const v2f*)(bp + 48 * D_IN + k);
        c0 = __builtin_amdgcn_wmma_f32_16x16x4_f32(false, a, false, b0, (short)0, c0, false, false);
        c1 = __builtin_amdgcn_wmma_f32_16x16x4_f32(false, a, false, b1, (short)0, c1, false, false);
        c2 = __builtin_amdgcn_wmma_f32_16x16x4_f32(false, a, false, b2, (short)0, c2, false, false);
        c3 = __builtin_amdgcn_wmma_f32_16x16x4_f32(false, a, false, b3, (short)0, c3, false, false);
    }

    // C/D layout: VGPR v, lane l -> M = v + 8*(l/16), N = l%16
    const int nlo = lane & 15;
    const int mhi = (lane >> 4) * 8;
    #pragma unroll
    for (int v = 0; v < 8; ++v) {
        size_t row = (size_t)(m0 + mhi + v) * H_DIM;
        out[row + n0 +  0 + nlo] = c0[v];
        out[row + n0 + 16 + nlo] = c1[v];
        out[row + n0 + 32 + nlo] = c2[v];
        out[row + n0 + 48 + nlo] = c3[v];
    }
}

// ---------------------------------------------------------------------------
// Edge scatter, layer 1: agg1[dst][:] += p1[src][:]   (128 floats/edge)
// 32 threads per edge, each handles a float4 chunk.
// ---------------------------------------------------------------------------
__global__ void sage_scatter1(const float* __restrict__ p1,
                              const long long* __restrict__ eidx, long long E,
                              float* __restrict__ agg1) {
    long long gid = (long long)blockIdx.x * blockDim.x + threadIdx.x;
    if (gid >= E * 32) return;
    long long e = gid >> 5;
    int c = (int)(gid & 31) * 4;
    long long s = eidx[e];
    long long d = eidx[E + e];
    float4 v = *(const float4*)(p1 + s * H_DIM + c);
    float* o = agg1 + d * H_DIM + c;
    unsafeAtomicAdd(o + 0, v.x);
    unsafeAtomicAdd(o + 1, v.y);
    unsafeAtomicAdd(o + 2, v.z);
    unsafeAtomicAdd(o + 3, v.w);
}

// ---------------------------------------------------------------------------
// h = relu(agg1/max(deg,1) + b1 + r1), written in place over agg1
// ---------------------------------------------------------------------------
__global__ void sage_hidden(float* __restrict__ agg1,
                            const float* __restrict__ r1,
                            const float* __restrict__ deg,
                            const float* __restrict__ b1) {
    int i = blockIdx.x * blockDim.x + threadIdx.x;
    if (i >= N_NODES * H_DIM) return;
    int n = i >> 7;          // /128
    int f = i & 127;
    float dg = fmaxf(deg[n], 1.0f);
    float v = agg1[i] / dg + b1[f] + r1[i];
    agg1[i] = fmaxf(v, 0.0f);
}

// ---------------------------------------------------------------------------
// GEMM2 (tiny, N=4): p2 = h @ W2_l^T, r2 = h @ W2_r^T. One node per thread.
// ---------------------------------------------------------------------------
__global__ __launch_bounds__(128)
void sage_gemm2(const float* __restrict__ h,
                const float* __restrict__ W2l,
                const float* __restrict__ W2r,
                float* __restrict__ p2,
                float* __restrict__ r2) {
    int n = blockIdx.x * blockDim.x + threadIdx.x;
    if (n >= N_NODES) return;
    const float4* hr = (const float4*)(h + (size_t)n * H_DIM);
    float accL[C_DIM] = {0.f, 0.f, 0.f, 0.f};
    float accR[C_DIM] = {0.f, 0.f, 0.f, 0.f};
    for (int k = 0; k < H_DIM / 4; ++k) {
        float4 hv = hr[k];
        #pragma unroll
        for (int c = 0; c < C_DIM; ++c) {
            float4 wl = ((const float4*)(W2l + c * H_DIM))[k];
            float4 wr = ((const float4*)(W2r + c * H_DIM))[k];
            accL[c] += hv.x * wl.x + hv.y * wl.y + hv.z * wl.z + hv.w * wl.w;
            accR[c] += hv.x * wr.x + hv.y * wr.y + hv.z * wr.z + hv.w * wr.w;
        }
    }
    #pragma unroll
    for (int c = 0; c < C_DIM; ++c) {
        p2[(size_t)n * C_DIM + c] = accL[c];
        r2[(size_t)n * C_DIM + c] = accR[c];
    }
}

// ---------------------------------------------------------------------------
// Edge scatter, layer 2: agg2[dst][:] += p2[src][:]   (4 floats/edge)
// ---------------------------------------------------------------------------
__global__ void sage_scatter2(const float* __restrict__ p2,
                              const long long* __restrict__ eidx, long long E,
                              float* __restrict__ agg2) {
    long long e = (long long)blockIdx.x * blockDim.x + threadIdx.x;
    if (e >= E) return;
    long long s = eidx[e];
    long long d = eidx[E + e];
    float4 v = *(const float4*)(p2 + s * C_DIM);
    float* o = agg2 + d * C_DIM;
    unsafeAtomicAdd(o + 0, v.x);
    unsafeAtomicAdd(o + 1, v.y);
    unsafeAtomicAdd(o + 2, v.z);
    unsafeAtomicAdd(o + 3, v.w);
}

// ---------------------------------------------------------------------------
// out = agg2/max(deg,1) + b2 + r2
// ---------------------------------------------------------------------------
__global__ void sage_out(const float* __restrict__ agg2,
                         const float* __restrict__ r2,
                         const float* __restrict__ deg,
                         const float* __restrict__ b2,
                         float* __restrict__ out) {
    int i = blockIdx.x * blockDim.x + threadIdx.x;
    if (i >= N_NODES * C_DIM) return;
    int n = i >> 2;
    int c = i & 3;
    float dg = fmaxf(deg[n], 1.0f);
    out[i] = agg2[i] / dg + b2[c] + r2[i];
}

extern "C" void kernel_launch(void* const* d_in, const int* in_sizes, int n_in,
                              void* d_out, int out_size, void* d_ws, size_t ws_size,
                              hipStream_t stream) {
    const float*     x    = (const float*)d_in[0];
    const long long* eidx = (const long long*)d_in[1];
    const float*     W1l  = (const float*)d_in[2];
    const float*     b1   = (const float*)d_in[3];
    const float*     W1r  = (const float*)d_in[4];
    const float*     W2l  = (const float*)d_in[5];
    const float*     b2   = (const float*)d_in[6];
    const float*     W2r  = (const float*)d_in[7];
    float*           out  = (float*)d_out;
    const long long  E    = (long long)in_sizes[1] / 2;

    // Workspace layout (floats). Zeroed block [deg|agg1|agg2] is contiguous.
    float* ws   = (float*)d_ws;
    float* deg  = ws;                                  // 50000
    float* agg1 = deg  + N_NODES;                      // 6.4M (becomes h)
    float* agg2 = agg1 + (size_t)N_NODES * H_DIM;      // 200000
    float* p1   = agg2 + (size_t)N_NODES * C_DIM;      // 6.4M
    float* r1   = p1   + (size_t)N_NODES * H_DIM;      // 6.4M
    float* p2   = r1   + (size_t)N_NODES * H_DIM;      // 200000
    float* r2   = p2   + (size_t)N_NODES * C_DIM;      // 200000

    size_t zero_floats = (size_t)N_NODES * (1 + H_DIM + C_DIM);
    hipMemsetAsync(ws, 0, zero_floats * sizeof(float), stream);

    // 1) degrees
    sage_deg_kernel<<<(unsigned)((E + 255) / 256), 256, 0, stream>>>(eidx, E, deg);

    // 2) WMMA projections: p1 = x W1_l^T, r1 = x W1_r^T
    sage_gemm1_wmma<<<N_NODES / 16, 128, 0, stream>>>(x, W1l, W1r, p1, r1);

    // 3) aggregate projected neighbors
    sage_scatter1<<<(unsigned)((E * 32 + 255) / 256), 256, 0, stream>>>(p1, eidx, E, agg1);

    // 4) h = relu(agg1/deg + b1 + r1)  (in place over agg1)
    sage_hidden<<<(N_NODES * H_DIM + 255) / 256, 256, 0, stream>>>(agg1, r1, deg, b1);

    // 5) tiny layer-2 projections
    sage_gemm2<<<(N_NODES + 127) / 128, 128, 0, stream>>>(agg1, W2l, W2r, p2, r2);

    // 6) aggregate layer-2 projections
    sage_scatter2<<<(unsigned)((E + 255) / 256), 256, 0, stream>>>(p2, eidx, E, agg2);

    // 7) final combine
    sage_out<<<(N_NODES * C_DIM + 255) / 256, 256, 0, stream>>>(agg2, r2, deg, b2, out);
}